// SCAP_SwiGLU_17772574671211
// MI455X (gfx1250) — compile-verified
//
#include <hip/hip_runtime.h>
#include <hip/hip_bf16.h>
#include <math.h>

// ---------------------------------------------------------------------------
// SCAP SwiGLU decode step, MI455X (gfx1250, wave32).
//   xm = where(|x|>0.5, x, 0)                             [4096]
//   up = xm @ Wupt        (skip inactive rows of Wupt)    [11008]
//   gate = silu(xm @ Wgatet)
//   gated = up * gate ; gm = where(|gated|>0.5, gated, 0)
//   out = gm @ Wdownt     (skip inactive rows of Wdownt)  [4096]
//
// Bandwidth-bound (541 MB fp32 dense -> ~23 us @ 23.3 TB/s; row-skipping cuts
// this to ~250 MB). Active weight rows are streamed once with NT loads and
// accumulated through V_WMMA_F32_16X16X4_F32 (fp32-exact, 16 cols x 4 rows
// per instruction). Compacted (idx,val) lists are zero-padded to a multiple
// of 128 so the hot loops are branch-free with int4/float4 uniform loads;
// the A operand is built with lane-mask arithmetic (no EXEC diamonds).
// ---------------------------------------------------------------------------

#define HIDDEN 4096
#define INTER  11008
#define TAU    0.5f
#define R1     16   // row-blocks for up/gate partials
#define R2     32   // row-blocks for down partials
#define PADN   128  // act-list padding granularity (= 4*R2, multiple of 4*R1)

typedef float v2f __attribute__((ext_vector_type(2)));
typedef float v8f __attribute__((ext_vector_type(8)));

// ---------------------------------------------------------------------------
// Deterministic stream compaction: (idx,val) of entries with |v| > tau,
// zero-padded up to the next multiple of PADN (zero val => no contribution).
// Single workgroup of 1024 threads (32 waves), block-wide scan per chunk.
// ---------------------------------------------------------------------------
__global__ void __launch_bounds__(1024)
scap_compact_kernel(const float* __restrict__ src, int n, float tau,
                    int* __restrict__ oidx, float* __restrict__ oval,
                    int* __restrict__ ocount)
{
    __shared__ int wsum[32];
    __shared__ int running;
    const int tid  = threadIdx.x;
    const int lane = tid & 31;
    const int wid  = tid >> 5;
    if (tid == 0) running = 0;
    __syncthreads();

    for (int base = 0; base < n; base += 1024) {
        const int i = base + tid;
        const float v = (i < n) ? src[i] : 0.0f;
        const bool flag = fabsf(v) > tau;

        const unsigned long long bal = __ballot(flag);      // wave32: low 32 bits
        const int pre  = __popcll(bal & ((1ull << lane) - 1ull));
        const int wtot = __popcll(bal);
        if (lane == 0) wsum[wid] = wtot;
        __syncthreads();

        int woff = 0, chunk = 0;
        for (int w = 0; w < 32; ++w) {
            if (w < wid) woff += wsum[w];
            chunk += wsum[w];
        }
        const int pos = running + woff + pre;
        __syncthreads();
        if (tid == 0) running += chunk;
        if (flag) { oidx[pos] = i; oval[pos] = v; }
        __syncthreads();
    }

    const int total  = running;
    const int totpad = (total + (PADN - 1)) & ~(PADN - 1);   // <= n (n % 128 == 0)
    for (int i = total + tid; i < totpad; i += 1024) {
        oidx[i] = 0;
        oval[i] = 0.0f;
    }
    if (tid == 0) *ocount = total;
}

// ---------------------------------------------------------------------------
// up/gate sparse GEMV via V_WMMA_F32_16X16X4_F32.
// grid = (86, R1), block = 256 (8 waves). Wave w owns 16-column tile
// blockIdx.x*8+w of BOTH Wupt and Wgatet; row-block blockIdx.y owns a
// contiguous chunk of the padded act list; 4 active rows per WMMA (K=4).
//
// A (16x4 f32): lane==M; lanes 0-15 hold K=0,1 ; lanes 16-31 hold K=2,3.
//   Only M=0 nonzero (the x values) -> built as m0*v + m16*v (no branches).
// B (4x16 f32): VGPR0: lanes 0-15 = K=0 row, lanes 16-31 = K=2 row;
//               VGPR1: lanes 0-15 = K=1 row, lanes 16-31 = K=3 row.
// C (16x16 f32): VGPR0, lanes 0-15 = row M=0 -> the 16 outputs.
// ---------------------------------------------------------------------------
__global__ void __launch_bounds__(256)
scap_upgate_wmma_kernel(const float* __restrict__ Wup,
                        const float* __restrict__ Wgate,
                        const int*   __restrict__ aidx,
                        const float* __restrict__ aval,
                        const int*   __restrict__ ncnt,
                        float* __restrict__ pu, float* __restrict__ pg)
{
    const int nact   = ncnt[0];
    const int npad   = (nact + 4 * R1 - 1) & ~(4 * R1 - 1);  // <= compaction pad
    const int chunk4 = npad / (4 * R1);                      // int4 groups / block

    const int lane = threadIdx.x & 31;
    const int wv   = threadIdx.x >> 5;
    const int tile = blockIdx.x * 8 + wv;      // 0..687
    const int n0   = tile * 16;
    const int rb   = blockIdx.y;               // 0..R1-1
    const unsigned colB = (unsigned)(n0 + (lane & 15)) * 4u; // byte offset in row
    const bool hi  = lane >= 16;
    const float m0  = (lane == 0)  ? 1.0f : 0.0f;   // hoisted lane masks
    const float m16 = (lane == 16) ? 1.0f : 0.0f;

    const int4*   ai4 = (const int4*)aidx;
    const float4* av4 = (const float4*)aval;
    const char* upB = (const char*)Wup;
    const char* gtB = (const char*)Wgate;

    v8f cu = {}; v8f cg = {};

#pragma unroll 4
    for (int t = 0; t < chunk4; ++t) {
        const int g = rb * chunk4 + t;         // uniform across wave
        const int4   h4 = ai4[g];              // one s_load_b128
        const float4 v4 = av4[g];              // one s_load_b128

        v2f a;                                  // flat VALU, no EXEC diamonds
        a.x = m0 * v4.x + m16 * v4.z;
        a.y = m0 * v4.y + m16 * v4.w;

        const unsigned offA = (unsigned)(hi ? h4.z : h4.x) * (INTER * 4u) + colB;
        const unsigned offB = (unsigned)(hi ? h4.w : h4.y) * (INTER * 4u) + colB;
        v2f bu, bg;
        bu.x = __builtin_nontemporal_load((const float*)(upB + offA));
        bu.y = __builtin_nontemporal_load((const float*)(upB + offB));
        bg.x = __builtin_nontemporal_load((const float*)(gtB + offA));
        bg.y = __builtin_nontemporal_load((const float*)(gtB + offB));

        cu = __builtin_amdgcn_wmma_f32_16x16x4_f32(false, a, false, bu,
                                                   (short)0, cu, false, false);
        cg = __builtin_amdgcn_wmma_f32_16x16x4_f32(false, a, false, bg,
                                                   (short)0, cg, false, false);
    }
    if (lane < 16) {
        pu[(size_t)rb * INTER + n0 + lane] = cu[0];
        pg[(size_t)rb * INTER + n0 + lane] = cg[0];
    }
}

// ---------------------------------------------------------------------------
// reduce partials, apply silu + gating; write dense gated vector [INTER].
// ---------------------------------------------------------------------------
__global__ void __launch_bounds__(256)
scap_reduce_upgate_kernel(const float* __restrict__ pu,
                          const float* __restrict__ pg,
                          float* __restrict__ gm)
{
    const int j = blockIdx.x * 256 + threadIdx.x;
    if (j >= INTER) return;
    float u = 0.0f, g = 0.0f;
#pragma unroll
    for (int r = 0; r < R1; ++r) {
        u += pu[(size_t)r * INTER + j];
        g += pg[(size_t)r * INTER + j];
    }
    const float s = g / (1.0f + expf(-g));   // silu(g) = g * sigmoid(g)
    gm[j] = u * s;                           // threshold applied by compaction
}

// ---------------------------------------------------------------------------
// down sparse GEMV via WMMA (same scheme). grid = (32, R2), block = 256.
// ---------------------------------------------------------------------------
__global__ void __launch_bounds__(256)
scap_down_wmma_kernel(const float* __restrict__ Wd,
                      const int*   __restrict__ aidx,
                      const float* __restrict__ aval,
                      const int*   __restrict__ ncnt,
                      float* __restrict__ pd)
{
    const int nact   = ncnt[0];
    const int npad   = (nact + 4 * R2 - 1) & ~(4 * R2 - 1);
    const int chunk4 = npad / (4 * R2);

    const int lane = threadIdx.x & 31;
    const int wv   = threadIdx.x >> 5;
    const int tile = blockIdx.x * 8 + wv;      // 0..255
    const int n0   = tile * 16;
    const int rb   = blockIdx.y;               // 0..R2-1
    const unsigned colB = (unsigned)(n0 + (lane & 15)) * 4u;
    const bool hi  = lane >= 16;
    const float m0  = (lane == 0)  ? 1.0f : 0.0f;
    const float m16 = (lane == 16) ? 1.0f : 0.0f;

    const int4*   ai4 = (const int4*)aidx;
    const float4* av4 = (const float4*)aval;
    const char* wdB = (const char*)Wd;

    v8f c = {};

#pragma unroll 4
    for (int t = 0; t < chunk4; ++t) {
        const int g = rb * chunk4 + t;
        const int4   h4 = ai4[g];
        const float4 v4 = av4[g];

        v2f a;
        a.x = m0 * v4.x + m16 * v4.z;
        a.y = m0 * v4.y + m16 * v4.w;

        const unsigned offA = (unsigned)(hi ? h4.z : h4.x) * (HIDDEN * 4u) + colB;
        const unsigned offB = (unsigned)(hi ? h4.w : h4.y) * (HIDDEN * 4u) + colB;
        v2f b;
        b.x = __builtin_nontemporal_load((const float*)(wdB + offA));
        b.y = __builtin_nontemporal_load((const float*)(wdB + offB));

        c = __builtin_amdgcn_wmma_f32_16x16x4_f32(false, a, false, b,
                                                  (short)0, c, false, false);
    }
    if (lane < 16) pd[(size_t)rb * HIDDEN + n0 + lane] = c[0];
}

__global__ void __launch_bounds__(256)
scap_reduce_down_kernel(const float* __restrict__ pd, float* __restrict__ out)
{
    const int j = blockIdx.x * 256 + threadIdx.x;
    if (j >= HIDDEN) return;
    float s = 0.0f;
#pragma unroll
    for (int r = 0; r < R2; ++r) s += pd[(size_t)r * HIDDEN + j];
    out[j] = s;
}

// ---------------------------------------------------------------------------
// launch: all on `stream`, deterministic, graph-capture safe.
// ---------------------------------------------------------------------------
extern "C" void kernel_launch(void* const* d_in, const int* in_sizes, int n_in,
                              void* d_out, int out_size, void* d_ws, size_t ws_size,
                              hipStream_t stream)
{
    const float* x      = (const float*)d_in[0];   // [4096]
    const float* Wupt   = (const float*)d_in[1];   // [4096, 11008]
    const float* Wgatet = (const float*)d_in[2];   // [4096, 11008]
    const float* Wdownt = (const float*)d_in[3];   // [11008, 4096]
    float*       out    = (float*)d_out;           // [4096]

    char* ws = (char*)d_ws;
    // 256B-aligned workspace layout (~2.0 MB total)
    int*   counts  = (int*)(ws + 0);                 // [0]=nact_x, [1]=nact_gm
    int*   a1_idx  = (int*)  (ws + 256);             // 4096 ints
    float* a1_val  = (float*)(ws + 256 + 16384);     // 4096 floats
    int*   a2_idx  = (int*)  (ws + 33024);           // 11008 ints
    float* a2_val  = (float*)(ws + 77056);           // 11008 floats
    float* gm      = (float*)(ws + 121088);          // 11008 floats
    float* pu      = (float*)(ws + 165120);          // R1*11008 floats
    float* pg      = (float*)(ws + 869632);          // R1*11008 floats
    float* pd      = (float*)(ws + 1574144);         // R2*4096 floats

    // 1) compact active x entries (|x| > 0.5), zero-padded to 128
    scap_compact_kernel<<<1, 1024, 0, stream>>>(x, HIDDEN, TAU, a1_idx, a1_val,
                                                counts + 0);
    // 2) up & gate partial GEMVs over active rows (WMMA f32 16x16x4)
    dim3 g2(86, R1);
    scap_upgate_wmma_kernel<<<g2, 256, 0, stream>>>(Wupt, Wgatet, a1_idx, a1_val,
                                                    counts + 0, pu, pg);
    // 3) reduce + silu + gate -> dense gated vector
    scap_reduce_upgate_kernel<<<43, 256, 0, stream>>>(pu, pg, gm);
    // 4) compact active gated entries (|gated| > 0.5), zero-padded to 128
    scap_compact_kernel<<<1, 1024, 0, stream>>>(gm, INTER, TAU, a2_idx, a2_val,
                                                counts + 1);
    // 5) down partial GEMV over active rows
    dim3 g5(32, R2);
    scap_down_wmma_kernel<<<g5, 256, 0, stream>>>(Wdownt, a2_idx, a2_val,
                                                  counts + 1, pd);
    // 6) final reduction -> out[4096]
    scap_reduce_down_kernel<<<16, 256, 0, stream>>>(pd, out);
}